// SpectralGCN_48318382080442
// MI455X (gfx1250) — compile-verified
//
#include <hip/hip_runtime.h>

#define N_NODES 8192
#define N_EDGES 262144
#define BATCH   2
#define C_IN    24      // NODE_FEATURES * NUM_TIMESTEPS
#define C_MID   32
#define C_OUT   12

typedef float v2f __attribute__((ext_vector_type(2)));
typedef float v8f __attribute__((ext_vector_type(8)));

// workspace layout in floats
#define WS_DEG  0                         // 8192
#define WS_Y    8192                      // 393216  (contiguous with DEG for one-shot zeroing)
#define WS_DINV 401408                    // 8192
#define WS_WN   409600                    // 262144
#define WS_Z1   671744                    // 393216
#define WS_Z2   1064960                   // 393216
#define WS_TOTAL 1458176

#define NC (BATCH * N_NODES * C_IN)       // 393216

// ---------------- sparse / elementwise stages ----------------

__global__ void sg_zero_kernel(float* p, int n) {
    int i = blockIdx.x * blockDim.x + threadIdx.x;
    if (i < n) p[i] = 0.0f;
}

__global__ void sg_deg_kernel(const int* __restrict__ ei,
                              const float* __restrict__ w,
                              float* __restrict__ deg) {
    int e = blockIdx.x * blockDim.x + threadIdx.x;
    if (e < N_EDGES) atomicAdd(&deg[ei[e]], w[e]);
}

__global__ void sg_dinv_kernel(const float* __restrict__ deg,
                               float* __restrict__ dinv) {
    int i = blockIdx.x * blockDim.x + threadIdx.x;
    if (i < N_NODES) dinv[i] = rsqrtf(deg[i]);
}

__global__ void sg_wn_kernel(const int* __restrict__ ei,
                             const float* __restrict__ w,
                             const float* __restrict__ dinv,
                             float* __restrict__ wn) {
    int e = blockIdx.x * blockDim.x + threadIdx.x;
    if (e < N_EDGES) {
        int src = ei[e];
        int dst = ei[N_EDGES + e];
        wn[e] = dinv[src] * w[e] * dinv[dst];
    }
}

// y[b,src,c] += wn[e] * v[b,dst,c]   (one thread per (edge, b, c))
__global__ void sg_spmm_kernel(const int* __restrict__ ei,
                               const float* __restrict__ wn,
                               const float* __restrict__ v,
                               float* __restrict__ y) {
    int tid = blockIdx.x * blockDim.x + threadIdx.x;
    if (tid >= N_EDGES * (BATCH * C_IN)) return;
    int e = tid / (BATCH * C_IN);
    int j = tid - e * (BATCH * C_IN);
    int b = j / C_IN;
    int c = j - b * C_IN;
    int src = ei[e];
    int dst = ei[N_EDGES + e];
    float val = wn[e] * v[(b * N_NODES + dst) * C_IN + c];
    atomicAdd(&y[(b * N_NODES + src) * C_IN + c], val);
}

// z1 = xf - y ; and re-zero y for the second SpMM pass
__global__ void sg_z1_kernel(const float* __restrict__ xf,
                             float* __restrict__ y,
                             float* __restrict__ z1) {
    int i = blockIdx.x * blockDim.x + threadIdx.x;
    if (i < NC) {
        z1[i] = xf[i] - y[i];
        y[i] = 0.0f;
    }
}

// z2 = 2*(z1 - y) - xf
__global__ void sg_z2_kernel(const float* __restrict__ xf,
                             const float* __restrict__ z1,
                             const float* __restrict__ y,
                             float* __restrict__ z2) {
    int i = blockIdx.x * blockDim.x + threadIdx.x;
    if (i < NC) {
        z2[i] = 2.0f * (z1[i] - y[i]) - xf[i];
    }
}

// ---------------- dense WMMA stage ----------------
// One wave per 16-node tile. GEMM1: [16 x 72] @ theta[72 x 32] (f32 WMMA 16x16x4),
// relu -> LDS, GEMM2: [16 x 32] @ W[32 x 12(pad16)] + b -> out.
__launch_bounds__(256)
__global__ void sg_gemm_kernel(const float* __restrict__ xf,
                               const float* __restrict__ z1,
                               const float* __restrict__ z2,
                               const float* __restrict__ theta,
                               const float* __restrict__ Wm,
                               const float* __restrict__ bias,
                               float* __restrict__ out) {
    __shared__ float hbuf[8][16 * 33];      // per-wave relu'd hidden tile, padded stride

    const int lane = threadIdx.x & 31;
    const int wave = threadIdx.x >> 5;
    const int tile = blockIdx.x * 8 + wave;          // 1024 tiles total
    const int bb   = tile >> 9;                      // tile / (N_NODES/16)
    const int nt   = tile & 511;
    const int node0 = nt * 16;
    const int m    = lane & 15;                      // A-row / B-col index
    const int half = lane >> 4;                      // selects K pair within k-step

    v8f acc0 = {0.f,0.f,0.f,0.f,0.f,0.f,0.f,0.f};
    v8f acc1 = {0.f,0.f,0.f,0.f,0.f,0.f,0.f,0.f};

    const float* zbufs[3] = { xf, z1, z2 };
    #pragma unroll
    for (int ph = 0; ph < 3; ++ph) {
        const float* zrow = zbufs[ph] + (size_t)(bb * N_NODES + node0 + m) * C_IN;
        #pragma unroll
        for (int kk = 0; kk < 6; ++kk) {
            const int k0 = 4 * kk + 2 * half;        // local channel in [0,24)
            v2f a;
            a.x = zrow[k0];
            a.y = zrow[k0 + 1];
            const int kg = ph * C_IN + k0;           // global K in [0,72)
            v2f b0, b1;
            b0.x = theta[(kg    ) * C_MID + m];
            b0.y = theta[(kg + 1) * C_MID + m];
            b1.x = theta[(kg    ) * C_MID + 16 + m];
            b1.y = theta[(kg + 1) * C_MID + 16 + m];
            acc0 = __builtin_amdgcn_wmma_f32_16x16x4_f32(false, a, false, b0,
                                                         (short)0, acc0, false, false);
            acc1 = __builtin_amdgcn_wmma_f32_16x16x4_f32(false, a, false, b1,
                                                         (short)0, acc1, false, false);
        }
    }

    // relu -> LDS (C/D layout: VGPR r holds row m = r + 8*half, col = lane&15)
    float* hw = hbuf[wave];
    #pragma unroll
    for (int r = 0; r < 8; ++r) {
        const int mr = r + 8 * half;
        hw[mr * 33 + m]      = fmaxf(acc0[r], 0.0f);
        hw[mr * 33 + 16 + m] = fmaxf(acc1[r], 0.0f);
    }
    __syncthreads();

    v8f acc = {0.f,0.f,0.f,0.f,0.f,0.f,0.f,0.f};
    const bool ovalid = (m < C_OUT);
    #pragma unroll
    for (int kk = 0; kk < 8; ++kk) {
        const int k0 = 4 * kk + 2 * half;            // K in [0,32)
        v2f a;
        a.x = hw[m * 33 + k0];
        a.y = hw[m * 33 + k0 + 1];
        v2f bw;
        bw.x = ovalid ? Wm[(k0    ) * C_OUT + m] : 0.0f;
        bw.y = ovalid ? Wm[(k0 + 1) * C_OUT + m] : 0.0f;
        acc = __builtin_amdgcn_wmma_f32_16x16x4_f32(false, a, false, bw,
                                                    (short)0, acc, false, false);
    }

    if (ovalid) {
        const float bo = bias[m];
        #pragma unroll
        for (int r = 0; r < 8; ++r) {
            const int mr = r + 8 * half;
            out[(size_t)(bb * N_NODES + node0 + mr) * C_OUT + m] = acc[r] + bo;
        }
    }
}

// ---------------- launch ----------------

extern "C" void kernel_launch(void* const* d_in, const int* in_sizes, int n_in,
                              void* d_out, int out_size, void* d_ws, size_t ws_size,
                              hipStream_t stream) {
    (void)in_sizes; (void)n_in; (void)out_size; (void)ws_size;

    const float* xf    = (const float*)d_in[0];   // [B,N,2,12] == [B,N,24] flat
    const int*   ei    = (const int*)  d_in[1];   // [2, N_EDGES]
    const float* ew    = (const float*)d_in[2];   // [N_EDGES]
    const float* theta = (const float*)d_in[3];   // [3,24,32]
    const float* Wm    = (const float*)d_in[4];   // [32,12]
    const float* bias  = (const float*)d_in[5];   // [12]
    float* out = (float*)d_out;

    float* ws   = (float*)d_ws;
    float* deg  = ws + WS_DEG;
    float* y    = ws + WS_Y;
    float* dinv = ws + WS_DINV;
    float* wn   = ws + WS_WN;
    float* zz1  = ws + WS_Z1;
    float* zz2  = ws + WS_Z2;

    const int T = 256;
    const int nZero = WS_DINV;               // deg + y contiguous
    sg_zero_kernel<<<(nZero + T - 1) / T, T, 0, stream>>>(ws, nZero);

    sg_deg_kernel <<<(N_EDGES + T - 1) / T, T, 0, stream>>>(ei, ew, deg);
    sg_dinv_kernel<<<(N_NODES + T - 1) / T, T, 0, stream>>>(deg, dinv);
    sg_wn_kernel  <<<(N_EDGES + T - 1) / T, T, 0, stream>>>(ei, ew, dinv, wn);

    const int nScat = N_EDGES * (BATCH * C_IN);
    sg_spmm_kernel<<<(nScat + T - 1) / T, T, 0, stream>>>(ei, wn, xf, y);   // y = Ln @ x
    sg_z1_kernel  <<<(NC + T - 1) / T, T, 0, stream>>>(xf, y, zz1);         // z1 = x - y; y = 0
    sg_spmm_kernel<<<(nScat + T - 1) / T, T, 0, stream>>>(ei, wn, zz1, y);  // y = Ln @ z1
    sg_z2_kernel  <<<(NC + T - 1) / T, T, 0, stream>>>(xf, zz1, y, zz2);    // z2 = 2(z1-y)-x

    const int nTiles = BATCH * N_NODES / 16;         // 1024 tiles, 8 waves/block
    sg_gemm_kernel<<<nTiles / 8, 256, 0, stream>>>(xf, zz1, zz2, theta, Wm, bias, out);
}